// BlockSequence_31147102830921
// MI455X (gfx1250) — compile-verified
//
#include <hip/hip_runtime.h>
#include <hip/hip_bf16.h>

// ---------------------------------------------------------------------------
// Shapes (compile-time constants from the reference)
// ---------------------------------------------------------------------------
#define NB    8
#define NPG   4096
#define NPTS  (NB * NPG)        // 32768
#define KNN   16
#define FIN   64
#define EMB   256
#define C0    32                // EMB / 2^(DEPTH+1)
#define NEGS  0.2f

typedef __attribute__((ext_vector_type(2))) float v2f;
typedef __attribute__((ext_vector_type(8))) float v8f;

__device__ __forceinline__ v8f wmma4(v2f a, v2f b, v8f c) {
  // D = A(16x4 f32) x B(4x16 f32) + C(16x16 f32)
  return __builtin_amdgcn_wmma_f32_16x16x4_f32(false, a, false, b,
                                               (short)0, c, false, false);
}

__device__ __forceinline__ float lrelu(float v) {
  return v >= 0.0f ? v : NEGS * v;
}

// ---------------------------------------------------------------------------
// Kernel 1: KNN — one thread per query point, register-resident sorted top-16,
// candidates streamed through a 256-point LDS tile.  Strict '<' everywhere
// reproduces jax.lax.top_k tie-breaking (lower index wins on equal distance).
// Emits GLOBAL index (batch_base + local) so later gathers are flat.
// ---------------------------------------------------------------------------
__global__ void knn_kernel(const float* __restrict__ pos,
                           int* __restrict__ idx_out) {
  __shared__ float sx[256], sy[256], sz[256];
  const int tid = threadIdx.x;
  const int n   = blockIdx.x * 256 + tid;      // global point id
  const int base = (n >> 12) << 12;            // batch start (NPG = 4096)

  const float px = pos[n * 3 + 0];
  const float py = pos[n * 3 + 1];
  const float pz = pos[n * 3 + 2];

  float bd[KNN];
  int   bi[KNN];
#pragma unroll
  for (int i = 0; i < KNN; ++i) { bd[i] = 3.0e38f; bi[i] = 0; }

  for (int t0 = 0; t0 < NPG; t0 += 256) {
    __syncthreads();
    const int c = base + t0 + tid;
    sx[tid] = pos[c * 3 + 0];
    sy[tid] = pos[c * 3 + 1];
    sz[tid] = pos[c * 3 + 2];
    __syncthreads();
    for (int j = 0; j < 256; ++j) {
      const float dx = px - sx[j], dy = py - sy[j], dz = pz - sz[j];
      const float d = dx * dx + dy * dy + dz * dz;
      if (d < bd[KNN - 1]) {
        bd[KNN - 1] = d;
        bi[KNN - 1] = base + t0 + j;
        // single bubble pass: list above slot 15 is already sorted
#pragma unroll
        for (int q = KNN - 1; q > 0; --q) {
          if (bd[q] < bd[q - 1]) {
            float td = bd[q]; bd[q] = bd[q - 1]; bd[q - 1] = td;
            int   ti = bi[q]; bi[q] = bi[q - 1]; bi[q - 1] = ti;
          }
        }
      }
    }
  }
#pragma unroll
  for (int i = 0; i < KNN; ++i) idx_out[n * KNN + i] = bi[i];
}

// ---------------------------------------------------------------------------
// Kernel 2: h0 = lrelu(x @ W_in + b_in)   (32768x64 @ 64x32)
// One wave per 16x16 output tile, f32 WMMA, K-loop of 16 steps.
// ---------------------------------------------------------------------------
__global__ void in_proj_kernel(const float* __restrict__ x,
                               const float* __restrict__ W,
                               const float* __restrict__ bias,
                               float* __restrict__ h0) {
  const int gw   = (blockIdx.x * blockDim.x + threadIdx.x) >> 5;
  const int lane = threadIdx.x & 31;
  const int colt = gw & 1;          // 32 / 16 = 2 column tiles
  const int rowt = gw >> 1;         // 2048 row tiles
  const int lm = lane & 15;
  const int lh = lane >> 4;         // 0: K pair {0,1}; 1: K pair {2,3}
  const int row = rowt * 16 + lm;
  const int col = colt * 16 + lm;

  v8f acc;
#pragma unroll
  for (int r = 0; r < 8; ++r) acc[r] = 0.0f;

  for (int kk = 0; kk < FIN / 4; ++kk) {
    const int k0 = kk * 4 + lh * 2;
    v2f a = *(const v2f*)(x + row * FIN + k0);      // 8B aligned (k0 even)
    v2f b;
    b.x = W[k0 * C0 + col];
    b.y = W[(k0 + 1) * C0 + col];
    acc = wmma4(a, b, acc);
  }

  const float bv = bias[col];
#pragma unroll
  for (int r = 0; r < 8; ++r) {
    const float v = lrelu(acc[r] + bv);
    h0[(rowt * 16 + r + lh * 8) * C0 + col] = v;
  }
}

// ---------------------------------------------------------------------------
// Kernel 3: EdgeConv aggregation.
//   feat[k] = [h_c, h[idx_k]-h_c, p[idx_k]-p_c]  (16 x F)
//   out     = max_k lrelu(feat @ W + b)          (O)
// One block per point; 16 rows of the WMMA tile == the 16 neighbors, so the
// k-max is an in-register reduce (8 acc VGPRs + one shfl_xor(16)).
// O/16 waves per block share one LDS feat tile; F padded to FP (mult of 4).
// ---------------------------------------------------------------------------
template <int C, int O>
__global__ void aggr_kernel(const float* __restrict__ hin,
                            const float* __restrict__ pos,
                            const int* __restrict__ idx,
                            const float* __restrict__ W,
                            const float* __restrict__ bias,
                            float* __restrict__ hout) {
  constexpr int F  = 2 * C + 3;
  constexpr int FP = F + 1;           // 68 / 132: multiple of 4, LDS stride
  constexpr int KT = FP / 4;          // WMMA K-steps
  constexpr int NW = O / 16;          // waves per block

  __shared__ float feat[16 * FP];
  __shared__ float s_hc[C];
  __shared__ float s_pc[3];
  __shared__ int   s_nb[KNN];

  const int n   = blockIdx.x;
  const int tid = threadIdx.x;

  if (tid < C)               s_hc[tid]         = hin[n * C + tid];
  else if (tid < C + 3)      s_pc[tid - C]     = pos[n * 3 + (tid - C)];
  else if (tid < C + 3 + KNN) s_nb[tid - C - 3] = idx[n * KNN + (tid - C - 3)];
  __syncthreads();

  // Cooperative feat build (gathers are L2-resident: whole problem < 100MB)
  for (int e = tid; e < 16 * F; e += NW * 32) {
    const int k = e / F;
    const int j = e - k * F;
    const int nb = s_nb[k];
    float v;
    if (j < C)            v = s_hc[j];
    else if (j < 2 * C)   v = hin[nb * C + (j - C)] - s_hc[j - C];
    else                  v = pos[nb * 3 + (j - 2 * C)] - s_pc[j - 2 * C];
    feat[k * FP + j] = v;
  }
  if (tid < 16) feat[tid * FP + F] = 0.0f;   // zero K-pad column
  __syncthreads();

  const int w    = tid >> 5;
  const int lane = tid & 31;
  const int lm = lane & 15;
  const int lh = lane >> 4;
  const int col = w * 16 + lm;

  v8f acc;
#pragma unroll
  for (int r = 0; r < 8; ++r) acc[r] = 0.0f;

  for (int kk = 0; kk < KT; ++kk) {
    const int k0 = kk * 4 + lh * 2;
    v2f a = *(const v2f*)&feat[lm * FP + k0];     // ds_load_b64
    v2f b;
    b.x = (k0 < F)     ? W[k0 * O + col]       : 0.0f;
    b.y = (k0 + 1 < F) ? W[(k0 + 1) * O + col] : 0.0f;
    acc = wmma4(a, b, acc);
  }

  const float bv = bias[col];
  float mx = -3.0e38f;
#pragma unroll
  for (int r = 0; r < 8; ++r) mx = fmaxf(mx, lrelu(acc[r] + bv));
  mx = fmaxf(mx, __shfl_xor(mx, 16));           // combine M 0-7 with M 8-15
  if (lane < 16) hout[n * O + col] = mx;
}

// ---------------------------------------------------------------------------
// Kernel 4: out = lrelu(h2 @ W_out + b_out + x @ W_sc + b_sc)
// Fused shortcut: two K-loops into one accumulator tile.
// ---------------------------------------------------------------------------
__global__ void out_proj_kernel(const float* __restrict__ h2,
                                const float* __restrict__ Wo,
                                const float* __restrict__ bo,
                                const float* __restrict__ x,
                                const float* __restrict__ Ws,
                                const float* __restrict__ bs,
                                float* __restrict__ out) {
  const int gw   = (blockIdx.x * blockDim.x + threadIdx.x) >> 5;
  const int lane = threadIdx.x & 31;
  const int colt = gw & 15;         // 256 / 16 = 16 column tiles
  const int rowt = gw >> 4;         // 2048 row tiles
  const int lm = lane & 15;
  const int lh = lane >> 4;
  const int row = rowt * 16 + lm;
  const int col = colt * 16 + lm;

  v8f acc;
#pragma unroll
  for (int r = 0; r < 8; ++r) acc[r] = 0.0f;

  for (int kk = 0; kk < 128 / 4; ++kk) {        // h2 @ W_out, K = 128
    const int k0 = kk * 4 + lh * 2;
    v2f a = *(const v2f*)(h2 + row * 128 + k0);
    v2f b;
    b.x = Wo[k0 * EMB + col];
    b.y = Wo[(k0 + 1) * EMB + col];
    acc = wmma4(a, b, acc);
  }
  for (int kk = 0; kk < FIN / 4; ++kk) {        // x @ W_sc, K = 64 (shortcut)
    const int k0 = kk * 4 + lh * 2;
    v2f a = *(const v2f*)(x + row * FIN + k0);
    v2f b;
    b.x = Ws[k0 * EMB + col];
    b.y = Ws[(k0 + 1) * EMB + col];
    acc = wmma4(a, b, acc);
  }

  const float bv = bo[col] + bs[col];
#pragma unroll
  for (int r = 0; r < 8; ++r) {
    const float v = lrelu(acc[r] + bv);
    out[(rowt * 16 + r + lh * 8) * EMB + col] = v;
  }
}

// ---------------------------------------------------------------------------
// Host launcher
// ---------------------------------------------------------------------------
extern "C" void kernel_launch(void* const* d_in, const int* in_sizes, int n_in,
                              void* d_out, int out_size, void* d_ws, size_t ws_size,
                              hipStream_t stream) {
  const float* x    = (const float*)d_in[0];
  const float* pos  = (const float*)d_in[1];
  const float* W_in = (const float*)d_in[2];
  const float* b_in = (const float*)d_in[3];
  const float* W_sc = (const float*)d_in[4];
  const float* b_sc = (const float*)d_in[5];
  const float* W_b0 = (const float*)d_in[6];
  const float* b_b0 = (const float*)d_in[7];
  const float* W_b1 = (const float*)d_in[8];
  const float* b_b1 = (const float*)d_in[9];
  const float* W_out = (const float*)d_in[10];
  const float* b_out = (const float*)d_in[11];
  float* out = (float*)d_out;

  // Workspace layout (30 MB total)
  char* ws = (char*)d_ws;
  int*   idxw = (int*)ws;                                   //  2 MB
  float* h0   = (float*)(ws + (size_t)NPTS * KNN * 4);      //  4 MB (N x 32)
  float* h1   = (float*)((char*)h0 + (size_t)NPTS * 32 * 4);//  8 MB (N x 64)
  float* h2   = (float*)((char*)h1 + (size_t)NPTS * 64 * 4);// 16 MB (N x 128)

  // 1. KNN indices (global rows)
  knn_kernel<<<NPTS / 256, 256, 0, stream>>>(pos, idxw);

  // 2. h0 = lrelu(x @ W_in + b_in): 2048x2 tiles, 4 waves/block
  in_proj_kernel<<<(2048 * 2) / 4, 128, 0, stream>>>(x, W_in, b_in, h0);

  // 3. EdgeConv block 0: C=32 -> O=64 (4 waves/point)
  aggr_kernel<32, 64><<<NPTS, 128, 0, stream>>>(h0, pos, idxw, W_b0, b_b0, h1);

  // 4. EdgeConv block 1: C=64 -> O=128 (8 waves/point)
  aggr_kernel<64, 128><<<NPTS, 256, 0, stream>>>(h1, pos, idxw, W_b1, b_b1, h2);

  // 5. out = lrelu(h2 @ W_out + x @ W_sc + b_out + b_sc): 2048x16 tiles
  out_proj_kernel<<<(2048 * 16) / 4, 128, 0, stream>>>(h2, W_out, b_out,
                                                       x, W_sc, b_sc, out);
}